// BinaryTreeTopDownLSTM_30434138259934
// MI455X (gfx1250) — compile-verified
//
#include <hip/hip_runtime.h>

// =====================================================================
// BinaryTreeTopDownLSTM for MI455X (gfx1250, wave32, WMMA)
//
// Key algebraic collapse: repeat(x,2) + position-independent composer
// means every tree level is constant across positions -> the whole tree
// is 10 composer steps on a (64,128) state. leaf(phl)/leaf(pcl) are
// per-batch-row constants broadcast over L=2048.
//
// Stage 1 (1 workgroup): 10 composer iterations + leaf(ph)/leaf(pc)
//   -> 64x512 broadcast table in d_ws. f32 WMMA 16x16x4, weights in LDS.
// Stage 2 (512 workgroups): embs @ [Wc|Wo] with f32 WMMA, fused
//   activation epilogue + broadcast-table fill. Bandwidth bound
//   (~470 MB traffic @ 23.3 TB/s), so stores are non-temporal.
// =====================================================================

typedef float v2f __attribute__((ext_vector_type(2)));
typedef float v8f __attribute__((ext_vector_type(8)));

__device__ __forceinline__ float sigmoidf_(float x) {
    return 1.0f / (1.0f + __expf(-x));
}

// D = A(16x4) * B(4x16) + C, all f32, wave32.
// A: lane%16 = M row, lane/16 selects K pair {0,1}/{2,3} in v0/v1.
// B: lane%16 = N col, lane/16 selects K pair, v0/v1 = K,K+1.
// C/D: VGPR v -> rows v (lanes 0-15) and v+8 (lanes 16-31), N = lane%16.
__device__ __forceinline__ v8f wmma_f32_k4(v2f a, v2f b, v8f c) {
    return __builtin_amdgcn_wmma_f32_16x16x4_f32(
        /*neg_a=*/false, a, /*neg_b=*/false, b,
        /*c_mod=*/(short)0, c, /*reuse_a=*/false, /*reuse_b=*/false);
}

#define V8ZERO {0.f,0.f,0.f,0.f,0.f,0.f,0.f,0.f}

// ---------------------------------------------------------------------
// Stage 1: tree (10 composer iterations on (64,128)) + leaf tables.
// One workgroup, 8 waves. Wave w owns output columns 16w..16w+15.
// Dynamic LDS: sph(8K f32) spc(8K) sW0(16K) sW1(16K) sW2(16K) = 256 KB.
// ---------------------------------------------------------------------
__global__ __launch_bounds__(256) void tree_leaf_kernel(
    const float* __restrict__ root_c, const float* __restrict__ root_h,
    const float* __restrict__ Wi, const float* __restrict__ bi,
    const float* __restrict__ Wf, const float* __restrict__ bf,
    const float* __restrict__ Wu, const float* __restrict__ bu,
    const float* __restrict__ Wc, const float* __restrict__ bc,
    const float* __restrict__ Wo, const float* __restrict__ bo,
    float* __restrict__ ws)
{
    extern __shared__ float smem[];
    float* sph = smem;              // 64*128
    float* spc = smem + 8192;       // 64*128
    float* sW0 = smem + 16384;      // 128*128 (Wi, later Wc)
    float* sW1 = smem + 32768;      // 128*128 (Wf, later Wo)
    float* sW2 = smem + 49152;      // 128*128 (Wu)

    const int tid = threadIdx.x;
    for (int idx = tid; idx < 8192; idx += 256) {
        spc[idx] = root_c[idx];
        sph[idx] = root_h[idx];
    }
    for (int idx = tid; idx < 16384; idx += 256) {
        sW0[idx] = Wi[idx];
        sW1[idx] = Wf[idx];
        sW2[idx] = Wu[idx];
    }
    __syncthreads();

    const int wave = tid >> 5;
    const int lane = tid & 31;
    const int l16  = lane & 15;
    const int hi   = lane >> 4;          // selects K pair {0,1} vs {2,3}
    const int col  = wave * 16 + l16;    // this wave's output column

    const float biv = bi[col], bfv = bf[col], buv = bu[col];

    float cr[4][8], hr[4][8];

    for (int it = 0; it < 10; ++it) {
        #pragma unroll
        for (int rt = 0; rt < 4; ++rt) {          // 4 row tiles of 16 -> 64 batch rows
            v8f ai = V8ZERO, af = V8ZERO, au = V8ZERO;
            const float* ph_row = sph + (rt * 16 + l16) * 128 + hi * 2;
            #pragma unroll
            for (int t = 0; t < 32; ++t) {        // K = 128 in steps of 4
                v2f a = *(const v2f*)(ph_row + 4 * t);
                const int k = 4 * t + hi * 2;
                v2f bI = { sW0[k * 128 + col], sW0[(k + 1) * 128 + col] };
                v2f bF = { sW1[k * 128 + col], sW1[(k + 1) * 128 + col] };
                v2f bU = { sW2[k * 128 + col], sW2[(k + 1) * 128 + col] };
                ai = wmma_f32_k4(a, bI, ai);
                af = wmma_f32_k4(a, bF, af);
                au = wmma_f32_k4(a, bU, au);
            }
            #pragma unroll
            for (int v = 0; v < 8; ++v) {
                const int row = rt * 16 + v + hi * 8;
                float iv = sigmoidf_(ai[v] + biv);
                float fv = sigmoidf_(af[v] + bfv);
                float uv = tanhf(au[v] + buv);
                float c  = iv * uv + fv * spc[row * 128 + col];
                cr[rt][v] = c;
                hr[rt][v] = tanhf(c);
            }
        }
        __syncthreads();   // all reads of sph/spc done before overwrite
        #pragma unroll
        for (int rt = 0; rt < 4; ++rt) {
            #pragma unroll
            for (int v = 0; v < 8; ++v) {
                const int row = rt * 16 + v + hi * 8;
                spc[row * 128 + col] = cr[rt][v];
                sph[row * 128 + col] = hr[rt][v];
            }
        }
        __syncthreads();
    }

    // Reuse weight LDS for Wc/Wo, then leaf(ph) and leaf(pc) -> d_ws table:
    // ws[b*512 + {0:cph, 128:cpc, 256:hph, 384:hpc} + col]
    for (int idx = tid; idx < 16384; idx += 256) {
        sW0[idx] = Wc[idx];
        sW1[idx] = Wo[idx];
    }
    __syncthreads();

    const float bcv = bc[col], bov = bo[col];
    for (int src = 0; src < 2; ++src) {
        const float* X = (src == 0) ? sph : spc;
        #pragma unroll
        for (int rt = 0; rt < 4; ++rt) {
            v8f ac = V8ZERO, ao = V8ZERO;
            const float* xrow = X + (rt * 16 + l16) * 128 + hi * 2;
            #pragma unroll
            for (int t = 0; t < 32; ++t) {
                v2f a = *(const v2f*)(xrow + 4 * t);
                const int k = 4 * t + hi * 2;
                v2f bC = { sW0[k * 128 + col], sW0[(k + 1) * 128 + col] };
                v2f bO = { sW1[k * 128 + col], sW1[(k + 1) * 128 + col] };
                ac = wmma_f32_k4(a, bC, ac);
                ao = wmma_f32_k4(a, bO, ao);
            }
            #pragma unroll
            for (int v = 0; v < 8; ++v) {
                const int b = rt * 16 + v + hi * 8;       // batch row
                float c = ac[v] + bcv;
                float h = sigmoidf_(ao[v] + bov) * tanhf(c);
                ws[b * 512 + src * 128 + col]       = c;  // cph / cpc
                ws[b * 512 + 256 + src * 128 + col] = h;  // hph / hpc
            }
        }
    }
}

// ---------------------------------------------------------------------
// Stage 2: leaf(embs) + output assembly.
// Rows r in [0,131072): b = r>>11, out row = [ce | cph | cpc | he | hph | hpc].
// 512 workgroups x 256 threads; each WG does 16 row-tiles (256 rows,
// always inside one batch since 2048 % 256 == 0).
// Wave w computes cols 16w..16w+15 of BOTH Wc and Wo products, and fills
// broadcast cols [64w,64w+64) of the 512-wide table region.
// ---------------------------------------------------------------------
#define TILES_PER_WG 16

__global__ __launch_bounds__(256) void leaf_main_kernel(
    const float* __restrict__ embs,
    const float* __restrict__ Wc, const float* __restrict__ bc,
    const float* __restrict__ Wo, const float* __restrict__ bo,
    const float* __restrict__ ws,
    float* __restrict__ out)
{
    const int tid  = threadIdx.x;
    const int wave = tid >> 5;
    const int lane = tid & 31;
    const int l16  = lane & 15;
    const int hi   = lane >> 4;
    const int col  = wave * 16 + l16;

    const int firstRow = blockIdx.x * (TILES_PER_WG * 16);

    // Hoist B operands for this wave's 16 columns of Wc and Wo: 128 VGPRs.
    v2f bcw[32], bow[32];
    #pragma unroll
    for (int t = 0; t < 32; ++t) {
        const int k = 4 * t + hi * 2;
        v2f x = { Wc[k * 128 + col], Wc[(k + 1) * 128 + col] };
        v2f y = { Wo[k * 128 + col], Wo[(k + 1) * 128 + col] };
        bcw[t] = x;
        bow[t] = y;
    }
    const float bcv = bc[col], bov = bo[col];

    // Per-batch broadcast slice: this wave covers table cols [64w, 64w+64).
    const int b  = firstRow >> 11;                   // row / 2048
    const int j0 = wave * 64 + lane * 2;             // 0..510, even
    v2f bcast = *(const v2f*)(ws + b * 512 + j0);
    // table col j -> out col: j<256 -> 128+j (cph,cpc) ; j>=256 -> 256+j (hph,hpc)
    const int outcol_b = j0 + ((wave < 4) ? 128 : 256);

    for (int rt = 0; rt < TILES_PER_WG; ++rt) {
        const int row0 = firstRow + rt * 16;
        v8f accc = V8ZERO, acco = V8ZERO;
        const float* arow = embs + (row0 + l16) * 128 + hi * 2;
        #pragma unroll
        for (int t = 0; t < 32; ++t) {
            v2f a = *(const v2f*)(arow + 4 * t);
            accc = wmma_f32_k4(a, bcw[t], accc);
            acco = wmma_f32_k4(a, bow[t], acco);
        }
        #pragma unroll
        for (int v = 0; v < 8; ++v) {
            const int row = row0 + v + hi * 8;
            float c = accc[v] + bcv;                     // ce
            float h = sigmoidf_(acco[v] + bov) * tanhf(c); // he
            __builtin_nontemporal_store(c, out + row * 768 + col);
            __builtin_nontemporal_store(h, out + row * 768 + 384 + col);
        }
        // Broadcast fill: 16 rows x this wave's 64 table columns.
        #pragma unroll
        for (int r = 0; r < 16; ++r) {
            v2f* p = (v2f*)(out + (row0 + r) * 768 + outcol_b);
            __builtin_nontemporal_store(bcast, p);
        }
    }
}

extern "C" void kernel_launch(void* const* d_in, const int* in_sizes, int n_in,
                              void* d_out, int out_size, void* d_ws, size_t ws_size,
                              hipStream_t stream) {
    const float* embs   = (const float*)d_in[0];
    const float* root_c = (const float*)d_in[1];
    const float* root_h = (const float*)d_in[2];
    const float* Wi = (const float*)d_in[3];
    const float* bi = (const float*)d_in[4];
    const float* Wf = (const float*)d_in[5];
    const float* bf = (const float*)d_in[6];
    const float* Wu = (const float*)d_in[7];
    const float* bu = (const float*)d_in[8];
    const float* Wc = (const float*)d_in[9];
    const float* bc = (const float*)d_in[10];
    const float* Wo = (const float*)d_in[11];
    const float* bo = (const float*)d_in[12];
    float* out = (float*)d_out;
    float* ws  = (float*)d_ws;   // needs 64*512*4 = 128 KB

    // Stage 1: tree + leaf tables (1 WGP, 256 KB dynamic LDS <= 320 KB).
    tree_leaf_kernel<<<1, 256, 65536 * sizeof(float), stream>>>(
        root_c, root_h, Wi, bi, Wf, bf, Wu, bu, Wc, bc, Wo, bo, ws);

    // Stage 2: bulk leaf(embs) + output assembly. 8192 row tiles / 16 per WG.
    leaf_main_kernel<<<512, 256, 0, stream>>>(embs, Wc, bc, Wo, bo, ws, out);
}